// RetinaNetDetector_12240656794133
// MI455X (gfx1250) — compile-verified
//
#include <hip/hip_runtime.h>
#include <hip/hip_bf16.h>
#include <stdint.h>

#define BATCH   16
#define NCLS    3
#define NANCH   9
#define NLEV    5
#define CAP     512           // per-level candidate cap (reference: top-1000)
#define MAXC    (NLEV * CAP)  // 2560 candidates max per image
#define DETS    300
#define NMSTHR  512

// -------- CDNA5 async global->LDS helpers (ASYNCcnt path) --------
__device__ __forceinline__ uint32_t lds_off(const void* p) {
  // For addrspace(3)-backed generic pointers, flat addr [31:0] == LDS byte offset.
  return (uint32_t)(uintptr_t)p;
}

__device__ __forceinline__ void async_g2l_b32(uint32_t dst_lds, const void* gsrc) {
  uint64_t ga = (uint64_t)(uintptr_t)gsrc;
  asm volatile("global_load_async_to_lds_b32 %0, %1, off"
               :: "v"(dst_lds), "v"(ga)
               : "memory");
}

__device__ __forceinline__ void wait_asynccnt0() {
  asm volatile("s_wait_asynccnt 0x0" ::: "memory");
}

// -------- kernel 0: zero the per-(image,level) counters --------
__global__ void zero_counts_kernel(int* __restrict__ counts) {
  int i = blockIdx.x * blockDim.x + threadIdx.x;
  if (i < BATCH * NLEV) counts[i] = 0;
}

// -------- kernel 1: sigmoid + threshold + decode + wave-aggregated compaction --------
__global__ void __launch_bounds__(256)
select_kernel(const float* __restrict__ cls,
              const float* __restrict__ reg,
              const float* __restrict__ anc,
              int fs, int lev,
              int* __restrict__ counts,
              float* __restrict__ cand)
{
  const int HW = fs * fs;
  int idx = blockIdx.x * blockDim.x + threadIdx.x;
  int total = BATCH * NANCH * NCLS * HW;
  if (idx >= total) return;

  // x fastest -> coalesced streaming read of the cls map
  int x  = idx % fs;
  int t  = idx / fs;
  int y  = t % fs;
  t     /= fs;
  int ch = t % (NANCH * NCLS);
  int b  = t / (NANCH * NCLS);

  float v = cls[((size_t)(b * (NANCH * NCLS) + ch) * fs + y) * fs + x];

  const float X_TH = -2.9444389791664403f;   // logit(0.05)
  bool pass = v > X_TH;

  // one atomic per wave32: ballot + popc compaction
  unsigned long long mask = __ballot(pass);
  int lane = threadIdx.x & 31;
  if (mask == 0ull) return;
  int cnt    = __popcll(mask);
  int leader = __ffsll(mask) - 1;
  int prefix = __popcll(mask & ((1ull << lane) - 1ull));
  int base = 0;
  if (pass && lane == leader)
    base = atomicAdd(&counts[b * NLEV + lev], cnt);
  base = __shfl(base, leader, 32);
  if (!pass) return;
  int slot = base + prefix;
  if (slot >= CAP) return;

  // sparse decode for survivors only
  int a = ch / NCLS;
  int c = ch - a * NCLS;

  float4 a4 = ((const float4*)anc)[(size_t)(y * fs + x) * NANCH + a];
  float wa  = a4.z - a4.x;
  float ha  = a4.w - a4.y;
  float cxa = a4.x + 0.5f * wa;
  float cya = a4.y + 0.5f * ha;

  size_t rbase = ((size_t)(b * (4 * NANCH) + a * 4) * fs + y) * fs + x;
  float dx = reg[rbase];
  float dy = reg[rbase + (size_t)HW];
  float dw = reg[rbase + (size_t)2 * HW];
  float dh = reg[rbase + (size_t)3 * HW];

  const float CLIPV = 4.1351665565490723f;   // log(1000/16)
  dw = fminf(dw, CLIPV);
  dh = fminf(dh, CLIPV);

  float cx = dx * wa + cxa;
  float cy = dy * ha + cya;
  float w  = __expf(dw) * wa;
  float h  = __expf(dh) * ha;

  float x1 = fminf(fmaxf(cx - 0.5f * w, 0.f), 1024.f);
  float y1 = fminf(fmaxf(cy - 0.5f * h, 0.f), 1024.f);
  float x2 = fminf(fmaxf(cx + 0.5f * w, 0.f), 1024.f);
  float y2 = fminf(fmaxf(cy + 0.5f * h, 0.f), 1024.f);

  float score = 1.f / (1.f + __expf(-v));

  // SoA: [B][NLEV][6][CAP] so per-field copies coalesce and async-stage cleanly
  float* cb = cand + (size_t)(b * NLEV + lev) * 6 * CAP;
  cb[0 * CAP + slot] = x1;
  cb[1 * CAP + slot] = y1;
  cb[2 * CAP + slot] = x2;
  cb[3 * CAP + slot] = y2;
  cb[4 * CAP + slot] = score;
  cb[5 * CAP + slot] = (float)c;
}

// -------- kernel 2: per-image NMS, candidates resident in LDS --------
__global__ void __launch_bounds__(NMSTHR)
nms_kernel(const float* __restrict__ cand,
           const int* __restrict__ counts,
           float* __restrict__ out)
{
  __shared__ float sx1[MAXC], sy1[MAXC], sx2[MAXC], sy2[MAXC], ssc[MAXC], slb[MAXC];
  __shared__ int   soff[NLEV + 1];
  __shared__ float red_s[NMSTHR / 32];
  __shared__ int   red_i[NMSTHR / 32];
  __shared__ float bcast[4];

  const int b   = blockIdx.x;
  const int tid = threadIdx.x;
  const int wid = tid >> 5;
  const int lane = tid & 31;

  if (tid == 0) {
    int acc = 0;
    for (int l = 0; l < NLEV; ++l) {
      soff[l] = acc;
      int c = counts[b * NLEV + l];
      acc += (c > CAP) ? CAP : c;
    }
    soff[NLEV] = acc;
  }
  __syncthreads();
  const int n = soff[NLEV];

  float* outb = out;
  float* outs = out + (size_t)BATCH * DETS * 4;
  float* outl = out + (size_t)BATCH * DETS * 5;

  if (n == 0) {
    for (int k = tid; k < DETS; k += NMSTHR) {
      outb[(size_t)(b * DETS + k) * 4 + 0] = 0.f;
      outb[(size_t)(b * DETS + k) * 4 + 1] = 0.f;
      outb[(size_t)(b * DETS + k) * 4 + 2] = 0.f;
      outb[(size_t)(b * DETS + k) * 4 + 3] = 0.f;
      outs[b * DETS + k] = 0.f;
      outl[b * DETS + k] = -1.f;
    }
    return;
  }

  // --- async-stage all candidates global -> LDS (6 fields per candidate) ---
  for (int i = tid; i < n; i += NMSTHR) {
    int l = 0;
    while (l < NLEV - 1 && i >= soff[l + 1]) ++l;
    int slot = i - soff[l];
    const float* g = cand + (size_t)(b * NLEV + l) * 6 * CAP + slot;
    async_g2l_b32(lds_off(&sx1[i]), g + 0 * CAP);
    async_g2l_b32(lds_off(&sy1[i]), g + 1 * CAP);
    async_g2l_b32(lds_off(&sx2[i]), g + 2 * CAP);
    async_g2l_b32(lds_off(&sy2[i]), g + 3 * CAP);
    async_g2l_b32(lds_off(&ssc[i]), g + 4 * CAP);
    async_g2l_b32(lds_off(&slb[i]), g + 5 * CAP);
  }
  wait_asynccnt0();
  __syncthreads();

  // per-class box offset (reference: boxes + label * 2*IMG on all 4 coords)
  for (int i = tid; i < n; i += NMSTHR) {
    float off = slb[i] * 2048.0f;
    sx1[i] += off; sy1[i] += off; sx2[i] += off; sy2[i] += off;
  }
  __syncthreads();

  for (int k = 0; k < DETS; ++k) {
    // ---- argmax over live scores (first-max tie-break like jnp.argmax) ----
    float best = -3.4e38f;
    int   bi   = 0x7FFFFFFF;
    for (int i = tid; i < n; i += NMSTHR) {
      float s = ssc[i];
      if (s > best || (s == best && i < bi)) { best = s; bi = i; }
    }
    for (int d = 16; d; d >>= 1) {
      float os = __shfl_xor(best, d, 32);
      int   oi = __shfl_xor(bi,   d, 32);
      if (os > best || (os == best && oi < bi)) { best = os; bi = oi; }
    }
    if (lane == 0) { red_s[wid] = best; red_i[wid] = bi; }
    __syncthreads();
    if (wid == 0) {
      const int nw = NMSTHR / 32;
      float bs  = (lane < nw) ? red_s[lane] : -3.4e38f;
      int   bix = (lane < nw) ? red_i[lane] : 0x7FFFFFFF;
      for (int d = 16; d; d >>= 1) {
        float os = __shfl_xor(bs,  d, 32);
        int   oi = __shfl_xor(bix, d, 32);
        if (os > bs || (os == bs && oi < bix)) { bs = os; bix = oi; }
      }
      if (lane == 0) {
        int j = bix;
        float sj   = ssc[j];
        float lab  = slb[j];
        float off  = lab * 2048.0f;
        bool valid = sj > 0.05f;
        outb[(size_t)(b * DETS + k) * 4 + 0] = valid ? sx1[j] - off : 0.f;
        outb[(size_t)(b * DETS + k) * 4 + 1] = valid ? sy1[j] - off : 0.f;
        outb[(size_t)(b * DETS + k) * 4 + 2] = valid ? sx2[j] - off : 0.f;
        outb[(size_t)(b * DETS + k) * 4 + 3] = valid ? sy2[j] - off : 0.f;
        outs[b * DETS + k] = valid ? sj  : 0.f;
        outl[b * DETS + k] = valid ? lab : -1.f;
        bcast[0] = sx1[j]; bcast[1] = sy1[j];
        bcast[2] = sx2[j]; bcast[3] = sy2[j];
        ssc[j] = -1.f;
      }
    }
    __syncthreads();

    // ---- suppression sweep (IoU on class-offset boxes) ----
    float bx1 = bcast[0], by1 = bcast[1], bx2 = bcast[2], by2 = bcast[3];
    float aj  = (bx2 - bx1) * (by2 - by1);
    for (int i = tid; i < n; i += NMSTHR) {
      float xx1 = fmaxf(bx1, sx1[i]);
      float yy1 = fmaxf(by1, sy1[i]);
      float xx2 = fminf(bx2, sx2[i]);
      float yy2 = fminf(by2, sy2[i]);
      float iw = fmaxf(xx2 - xx1, 0.f);
      float ih = fmaxf(yy2 - yy1, 0.f);
      float inter = iw * ih;
      float ai = (sx2[i] - sx1[i]) * (sy2[i] - sy1[i]);
      float iou = inter / (aj + ai - inter + 1e-7f);
      if (iou > 0.5f) ssc[i] = -1.f;
    }
    __syncthreads();
  }
}

extern "C" void kernel_launch(void* const* d_in, const int* in_sizes, int n_in,
                              void* d_out, int out_size, void* d_ws, size_t ws_size,
                              hipStream_t stream) {
  (void)in_sizes; (void)n_in; (void)out_size; (void)ws_size;
  static const int FS[NLEV] = {128, 64, 32, 16, 8};

  int*   counts = (int*)d_ws;
  float* cand   = (float*)((char*)d_ws + 512);   // [B][5][6][CAP] floats (~960 KB)

  zero_counts_kernel<<<1, 128, 0, stream>>>(counts);

  // setup_inputs dict order: cls_l i, reg_l i, anchors_l i interleaved per level
  for (int l = 0; l < NLEV; ++l) {
    const float* cls = (const float*)d_in[3 * l + 0];
    const float* reg = (const float*)d_in[3 * l + 1];
    const float* anc = (const float*)d_in[3 * l + 2];
    int fs = FS[l];
    int total = BATCH * NANCH * NCLS * fs * fs;
    select_kernel<<<(total + 255) / 256, 256, 0, stream>>>(
        cls, reg, anc, fs, l, counts, cand);
  }

  nms_kernel<<<BATCH, NMSTHR, 0, stream>>>(cand, counts, (float*)d_out);
}